// GAT_40913858462401
// MI455X (gfx1250) — compile-verified
//
#include <hip/hip_runtime.h>
#include <hip/hip_bf16.h>

typedef float v2f __attribute__((ext_vector_type(2)));
typedef float v8f __attribute__((ext_vector_type(8)));

#define F_IN 128
#define F_E  16
#define NH   8
#define NC   32
#define HC   256   // NH*NC
#define NEG_SLOPE 0.2f

// ---------- monotone float <-> uint key for atomic segment-max ----------
__device__ __forceinline__ unsigned ordkey(float f) {
    unsigned u = __float_as_uint(f);
    return (u & 0x80000000u) ? ~u : (u | 0x80000000u);
}
__device__ __forceinline__ float ordinv(unsigned k) {
    unsigned u = (k & 0x80000000u) ? (k & 0x7fffffffu) : ~k;
    return __uint_as_float(u);
}

// ---------- 0) zero accumulation regions ----------
__global__ void GAT_zero(float* p, long n) {
    long i = (long)blockIdx.x * blockDim.x + threadIdx.x;
    long stride = (long)gridDim.x * blockDim.x;
    for (; i < n; i += stride) p[i] = 0.0f;
}

// ---------- 1) deg + loop-attr sums (one thread per (edge, f)) ----------
__global__ void GAT_degloop(const int* __restrict__ dst,
                            const float* __restrict__ eattr,
                            float* __restrict__ deg,
                            float* __restrict__ loop, int E) {
    long idx = (long)blockIdx.x * blockDim.x + threadIdx.x;
    long e = idx >> 4;
    int  f = (int)(idx & 15);
    if (e >= E) return;
    int d = dst[e];
    atomicAdd(&loop[(long)d * F_E + f], eattr[e * F_E + f]);
    if (f == 0) atomicAdd(&deg[d], 1.0f);
}

// ---------- 2) loop_attr = sum / max(deg,1) ----------
__global__ void GAT_loopdiv(float* __restrict__ loop,
                            const float* __restrict__ deg, int N) {
    long idx = (long)blockIdx.x * blockDim.x + threadIdx.x;
    if (idx >= (long)N * F_E) return;
    int n = (int)(idx >> 4);
    loop[idx] = loop[idx] / fmaxf(deg[n], 1.0f);
}

// ---------- 3) M[f][h] = sum_c W_edge[h*C+c][f] * att_edge[h][c] ----------
__global__ void GAT_foldM(const float* __restrict__ We,
                          const float* __restrict__ att_edge,
                          float* __restrict__ M) {
    int tid = threadIdx.x;           // 128 threads
    if (tid >= F_E * NH) return;
    int f = tid >> 3, h = tid & 7;
    float s = 0.0f;
    for (int c = 0; c < NC; ++c)
        s += We[(long)(h * NC + c) * F_E + f] * att_edge[h * NC + c];
    M[f * NH + h] = s;
}

// ---------- 4) xh = x @ W^T via V_WMMA_F32_16X16X4_F32 ----------
// One wave computes a 16x64 output strip (4 accumulators) so the A fragment
// is reused across 4 WMMAs per k-step. All loads are unconditional b64:
// tail rows are handled by CLAMPING the A row index (branch-free, keeps EXEC
// all-ones around WMMA as the ISA requires); stores are guarded.
// 32-bit A layout: lane l<16 -> M=l,K={0,1}; lane l>=16 -> M=l-16,K={2,3}
// 32-bit B layout: VGPR0 = rows K=0 (lanes 0-15)/K=2 (lanes 16-31), VGPR1 = K=1/K=3
// C/D layout: vgpr v, lane half -> row v + 8*half, col = lane&15
__global__ void GAT_gemm_xh(const float* __restrict__ x,
                            const float* __restrict__ W,
                            float* __restrict__ xh, int N) {
    int wave = threadIdx.x >> 5;
    int lane = threadIdx.x & 31;
    int half = lane >> 4;
    int l    = lane & 15;
    long rowTiles = ((long)N + 15) >> 4;
    long wtile = (long)blockIdx.x * 8 + wave;       // 8 waves / block, 16x64 each
    if (wtile >= rowTiles * 4) return;              // wave-uniform guard
    long rowTile = wtile >> 2;                      // 4 column groups of 64
    int  colGrp  = (int)(wtile & 3);
    long row0 = rowTile * 16;
    int  col0 = colGrp * 64;

    long arow = row0 + l;
    if (arow >= N) arow = N - 1;                    // clamp: loads stay unconditional
    const float* xp  = x + arow * F_IN + 2 * half;
    const float* wp0 = W + (long)(col0 +  0 + l) * F_IN + 2 * half;
    const float* wp1 = W + (long)(col0 + 16 + l) * F_IN + 2 * half;
    const float* wp2 = W + (long)(col0 + 32 + l) * F_IN + 2 * half;
    const float* wp3 = W + (long)(col0 + 48 + l) * F_IN + 2 * half;

    v8f a0 = {0.f,0.f,0.f,0.f,0.f,0.f,0.f,0.f};
    v8f a1 = a0, a2 = a0, a3 = a0;
    #pragma unroll
    for (int k0 = 0; k0 < F_IN; k0 += 4) {
        v2f a  = *(const v2f*)(xp  + k0);
        v2f b0 = *(const v2f*)(wp0 + k0);
        v2f b1 = *(const v2f*)(wp1 + k0);
        v2f b2 = *(const v2f*)(wp2 + k0);
        v2f b3 = *(const v2f*)(wp3 + k0);
        a0 = __builtin_amdgcn_wmma_f32_16x16x4_f32(false, a, false, b0, (short)0, a0, false, false);
        a1 = __builtin_amdgcn_wmma_f32_16x16x4_f32(false, a, false, b1, (short)0, a1, false, false);
        a2 = __builtin_amdgcn_wmma_f32_16x16x4_f32(false, a, false, b2, (short)0, a2, false, false);
        a3 = __builtin_amdgcn_wmma_f32_16x16x4_f32(false, a, false, b3, (short)0, a3, false, false);
    }
    #pragma unroll
    for (int v = 0; v < 8; ++v) {
        long r = row0 + v + 8 * half;
        if (r < N) {
            float* o = xh + r * HC + col0 + l;
            o[0]  = a0[v];
            o[16] = a1[v];
            o[32] = a2[v];
            o[48] = a3[v];
        }
    }
}

// ---------- 5) per-node attention terms ----------
__global__ void GAT_nodeatt(const float* __restrict__ xh,
                            const float* __restrict__ att_src,
                            const float* __restrict__ att_dst,
                            float* __restrict__ asrc,
                            float* __restrict__ adst, int N) {
    long idx = (long)blockIdx.x * blockDim.x + threadIdx.x;
    if (idx >= (long)N * NH) return;
    long n = idx >> 3;
    int  h = (int)(idx & 7);
    const float* xr = xh + n * HC + h * NC;
    float ss = 0.0f, sd = 0.0f;
    for (int c = 0; c < NC; ++c) {
        float v = xr[c];
        ss += v * att_src[h * NC + c];
        sd += v * att_dst[h * NC + c];
    }
    asrc[idx] = ss;
    adst[idx] = sd;
}

// ---------- 6) alpha + leaky relu + segment max (edges incl. self loops) ----------
__global__ void GAT_alpha(const int* __restrict__ src, const int* __restrict__ dst,
                          const float* __restrict__ eattr, const float* __restrict__ loop,
                          const float* __restrict__ M,
                          const float* __restrict__ asrc, const float* __restrict__ adst,
                          float* __restrict__ exbuf, unsigned* __restrict__ amax,
                          int E, int N) {
    long t = (long)blockIdx.x * blockDim.x + threadIdx.x;
    long T = (long)E + N;
    if (t >= T) return;
    int s, d;
    const float* row;
    if (t < E) { s = src[t]; d = dst[t]; row = eattr + t * F_E; }
    else       { s = (int)(t - E); d = s;  row = loop + (long)s * F_E; }
    float ea[F_E];
    #pragma unroll
    for (int f = 0; f < F_E; ++f) ea[f] = row[f];
    #pragma unroll
    for (int h = 0; h < NH; ++h) {
        float ae = 0.0f;
        #pragma unroll
        for (int f = 0; f < F_E; ++f) ae += ea[f] * M[f * NH + h];
        float al = asrc[(long)s * NH + h] + adst[(long)d * NH + h] + ae;
        al = (al >= 0.0f) ? al : NEG_SLOPE * al;
        exbuf[t * NH + h] = al;
        atomicMax(&amax[(long)d * NH + h], ordkey(al));
    }
}

// ---------- 7) exp + denom ----------
__global__ void GAT_expsum(const int* __restrict__ dst,
                           float* __restrict__ exbuf,
                           const unsigned* __restrict__ amax,
                           float* __restrict__ denom, int E, int N) {
    long idx = (long)blockIdx.x * blockDim.x + threadIdx.x;
    long T = ((long)E + N) * NH;
    if (idx >= T) return;
    long t = idx >> 3;
    int  h = (int)(idx & 7);
    int d = (t < E) ? dst[t] : (int)(t - E);
    float m = ordinv(amax[(long)d * NH + h]);
    float e = __expf(exbuf[idx] - m);
    exbuf[idx] = e;
    atomicAdd(&denom[(long)d * NH + h], e);
}

// ---------- 8) weighted message scatter (block per edge, 256 threads = H*C) ----------
__global__ void GAT_scatter(const int* __restrict__ src, const int* __restrict__ dst,
                            const float* __restrict__ xh,
                            const float* __restrict__ exbuf,
                            const float* __restrict__ denom,
                            float* __restrict__ accum, int E, int N) {
    long t = blockIdx.x;
    int tid = threadIdx.x;           // 0..255 -> h = tid/32, c = tid%32
    int s, d;
    if (t < E) { s = src[t]; d = dst[t]; }
    else       { s = (int)(t - E); d = s; }
    int h = tid >> 5;
    float att = exbuf[t * NH + h] / denom[(long)d * NH + h];
    float val = xh[(long)s * HC + tid] * att;
    atomicAdd(&accum[(long)d * HC + tid], val);
}

// ---------- 9) head mean + bias ----------
__global__ void GAT_final(const float* __restrict__ accum,
                          const float* __restrict__ bias,
                          float* __restrict__ out, int N) {
    long idx = (long)blockIdx.x * blockDim.x + threadIdx.x;
    if (idx >= (long)N * NC) return;
    long n = idx >> 5;
    int  c = (int)(idx & 31);
    float sum = 0.0f;
    #pragma unroll
    for (int h = 0; h < NH; ++h) sum += accum[n * HC + h * NC + c];
    out[idx] = sum * (1.0f / NH) + bias[c];
}

extern "C" void kernel_launch(void* const* d_in, const int* in_sizes, int n_in,
                              void* d_out, int out_size, void* d_ws, size_t ws_size,
                              hipStream_t stream) {
    const float* x        = (const float*)d_in[0];
    const int*   eidx     = (const int*)  d_in[1];
    const float* eattr    = (const float*)d_in[2];
    const float* W        = (const float*)d_in[3];
    const float* We       = (const float*)d_in[4];
    const float* att_src  = (const float*)d_in[5];
    const float* att_dst  = (const float*)d_in[6];
    const float* att_edge = (const float*)d_in[7];
    const float* bias     = (const float*)d_in[8];

    const int N = in_sizes[0] / F_IN;
    const int E = in_sizes[1] / 2;
    const int* src = eidx;
    const int* dst = eidx + E;

    // ---- workspace layout (floats) ----
    float* ws = (float*)d_ws;
    const long Nl = N, El = E, Tl = El + Nl;
    float*    w_deg   = ws;                         // N
    float*    w_loop  = ws + Nl;                    // 16N
    float*    w_denom = ws + 17 * Nl;               // 8N
    unsigned* w_amax  = (unsigned*)(ws + 25 * Nl);  // 8N (ordered keys; 0 == min)
    float*    w_acc   = ws + 33 * Nl;               // 256N
    const long zeroSpan = 289 * Nl;                 // everything above
    float*    w_xh    = ws + zeroSpan;              // 256N
    float*    w_asrc  = ws + zeroSpan + 256 * Nl;   // 8N
    float*    w_adst  = ws + zeroSpan + 264 * Nl;   // 8N
    float*    w_M     = ws + zeroSpan + 272 * Nl;   // 128
    float*    w_ex    = ws + zeroSpan + 272 * Nl + 128;  // 8*(E+N)

    const int B = 256;
    // 0) zero accumulators (deg, loop sums, denom, amax keys, HC accumulator)
    GAT_zero<<<2048, B, 0, stream>>>(ws, zeroSpan);
    // 1) degree + loop attr sums
    {
        long th = El * F_E;
        GAT_degloop<<<(int)((th + B - 1) / B), B, 0, stream>>>(dst, eattr, w_deg, w_loop, E);
    }
    // 2) mean
    {
        long th = Nl * F_E;
        GAT_loopdiv<<<(int)((th + B - 1) / B), B, 0, stream>>>(w_loop, w_deg, N);
    }
    // 3) fold edge attention matrix
    GAT_foldM<<<1, 128, 0, stream>>>(We, att_edge, w_M);
    // 4) projection GEMM (WMMA f32 16x16x4), 8 waves/block, 16x64 strip per wave
    {
        long rowTiles = (Nl + 15) >> 4;
        long wtiles = rowTiles * 4;                 // 4 column groups of 64
        GAT_gemm_xh<<<(int)((wtiles + 7) / 8), B, 0, stream>>>(x, W, w_xh, N);
    }
    // 5) per-node attention terms
    {
        long th = Nl * NH;
        GAT_nodeatt<<<(int)((th + B - 1) / B), B, 0, stream>>>(w_xh, att_src, att_dst,
                                                               w_asrc, w_adst, N);
    }
    // 6) alpha + segment max
    GAT_alpha<<<(int)((Tl + B - 1) / B), B, 0, stream>>>(src, dst, eattr, w_loop, w_M,
                                                         w_asrc, w_adst, w_ex, w_amax, E, N);
    // 7) exp + denom
    {
        long th = Tl * NH;
        GAT_expsum<<<(int)((th + B - 1) / B), B, 0, stream>>>(dst, w_ex, w_amax, w_denom, E, N);
    }
    // 8) weighted scatter-sum (one block per edge incl. self loops)
    GAT_scatter<<<(int)Tl, B, 0, stream>>>(src, dst, w_xh, w_ex, w_denom, w_acc, E, N);
    // 9) head mean + bias
    {
        long th = Nl * NC;
        GAT_final<<<(int)((th + B - 1) / B), B, 0, stream>>>(w_acc, bias, (float*)d_out, N);
    }
}